// GraphSAGE_34196529610765
// MI455X (gfx1250) — compile-verified
//
#include <hip/hip_runtime.h>

#define N_NODES 100000
#define N_EDGES 1600000
#define IN_C 128
#define HID_C 128
#define OUT_C 64

typedef __attribute__((ext_vector_type(2))) float v2f;
typedef __attribute__((ext_vector_type(8))) float v8f;

// ---------------------------------------------------------------------------
// Native CDNA5 float atomic max (no return), device scope.
// Guarantees GLOBAL_ATOMIC_MAX_NUM_F32 instead of a possible CAS loop.
// ---------------------------------------------------------------------------
__device__ __forceinline__ void atomic_max_f32_dev(float* p, float v) {
  asm volatile("global_atomic_max_num_f32 %0, %1, off scope:SCOPE_DEV"
               :: "v"(p), "v"(v)
               : "memory");
}

// ---------------------------------------------------------------------------
// Fill a float buffer with -inf (float4 vectorized)
// ---------------------------------------------------------------------------
__global__ void fill_neginf_kernel(float4* __restrict__ p, int n4) {
  int i = blockIdx.x * blockDim.x + threadIdx.x;
  if (i < n4) {
    const float ninf = -__builtin_inff();
    p[i] = make_float4(ninf, ninf, ninf, ninf);
  }
}

// ---------------------------------------------------------------------------
// Scatter-max: one WAVE per edge. 32 lanes x float4 = 128 channels exactly.
// Edge index e is wave-uniform -> src/dst become scalar loads (s_load),
// cutting index traffic 32x vs per-thread mapping. Gather is a coalesced
// 512B float4 read; reduce is 4 native f32 max atomics per lane.
// E = 1.6M, 8 waves/block -> 200000 blocks exactly (no guards).
// ---------------------------------------------------------------------------
__global__ __launch_bounds__(256)
void scatter_max_kernel(const float* __restrict__ feat,
                        const long long* __restrict__ src,
                        const long long* __restrict__ dst,
                        float* __restrict__ aggr) {
  const int wave = threadIdx.x >> 5;
  const int lane = threadIdx.x & 31;
  const int e = blockIdx.x * 8 + wave;          // wave-uniform
  const int s = (int)src[e];
  const int d = (int)dst[e];
  const float4 v = *(const float4*)(feat + (size_t)s * 128 + lane * 4);
  float* o = aggr + (size_t)d * 128 + lane * 4;
  atomic_max_f32_dev(o + 0, v.x);
  atomic_max_f32_dev(o + 1, v.y);
  atomic_max_f32_dev(o + 2, v.z);
  atomic_max_f32_dev(o + 3, v.w);
}

// ---------------------------------------------------------------------------
// Fused SAGE layer GEMM using V_WMMA_F32_16X16X4_F32 (exact f32):
//   out[m][n] = act( fix(aggr)[m][:] @ Wl[:, n] + bias[n] + xin[m][:] @ Wr[:, n] )
// fix() replaces -inf (isolated nodes) with 0.
// One wave computes one 16x16 output tile; K fully unrolled (K,N constexpr).
// M = 100000 = 6250*16 exactly -> no tail guards -> EXEC all-1s for WMMA.
//
// A layout (16x4 f32): lane<16 -> row=lane, holds K=k0,k0+1 in a.x,a.y
//                      lane>=16 -> row=lane-16, holds K=k0+2,k0+3
// B layout (4x16 f32): vgpr j, lane<16 -> B[k0+j][n0+lane]; lane>=16 -> B[k0+2+j][..]
// C/D layout: acc[i] = out[m0 + i + 8*(lane>>4)][n0 + (lane&15)]
// ---------------------------------------------------------------------------
template <int K, int N, bool RELU>
__global__ __launch_bounds__(256)
void sage_gemm_kernel(const float* __restrict__ aggr,
                      const float* __restrict__ xin,
                      const float* __restrict__ Wl,
                      const float* __restrict__ bias,
                      const float* __restrict__ Wr,
                      float* __restrict__ out) {
  const int lane  = threadIdx.x & 31;
  const int wave  = threadIdx.x >> 5;
  const int m0    = blockIdx.x * 16;
  const int n0    = wave * 16;
  const int half  = lane >> 4;      // 0 or 1
  const int idx16 = lane & 15;

  const float ninf = -__builtin_inff();

  const float* arow = aggr + (size_t)(m0 + idx16) * K;
  const float* xrow = xin  + (size_t)(m0 + idx16) * K;
  const int colB = n0 + idx16;

  v8f acc = {};

#pragma unroll
  for (int k0 = 0; k0 < K; k0 += 4) {
    const int kb = k0 + 2 * half;

    // A = fix(aggr) tile (contiguous float2 per lane)
    float a0 = arow[kb];
    float a1 = arow[kb + 1];
    a0 = (a0 == ninf) ? 0.0f : a0;
    a1 = (a1 == ninf) ? 0.0f : a1;
    v2f a; a.x = a0; a.y = a1;

    // B = Wl tile
    v2f b;
    b.x = Wl[(size_t)kb * N + colB];
    b.y = Wl[(size_t)(kb + 1) * N + colB];

    acc = __builtin_amdgcn_wmma_f32_16x16x4_f32(
        false, a, false, b, (short)0, acc, false, false);

    // A2 = xin tile (root path)
    v2f a2; a2.x = xrow[kb]; a2.y = xrow[kb + 1];

    // B2 = Wr tile
    v2f b2;
    b2.x = Wr[(size_t)kb * N + colB];
    b2.y = Wr[(size_t)(kb + 1) * N + colB];

    acc = __builtin_amdgcn_wmma_f32_16x16x4_f32(
        false, a2, false, b2, (short)0, acc, false, false);
  }

  const float bv = bias[colB];
#pragma unroll
  for (int i = 0; i < 8; ++i) {
    float v = acc[i] + bv;
    if (RELU) v = fmaxf(v, 0.0f);
    out[(size_t)(m0 + i + 8 * half) * N + colB] = v;
  }
}

// ---------------------------------------------------------------------------
// Launch
// ---------------------------------------------------------------------------
extern "C" void kernel_launch(void* const* d_in, const int* in_sizes, int n_in,
                              void* d_out, int out_size, void* d_ws, size_t ws_size,
                              hipStream_t stream) {
  const float*     x   = (const float*)d_in[0];
  const long long* ei  = (const long long*)d_in[1];
  const float*     W1l = (const float*)d_in[2];
  const float*     b1l = (const float*)d_in[3];
  const float*     W1r = (const float*)d_in[4];
  const float*     W2l = (const float*)d_in[5];
  const float*     b2l = (const float*)d_in[6];
  const float*     W2r = (const float*)d_in[7];
  float*           out = (float*)d_out;

  const long long* src = ei;              // edge_index[0]
  const long long* dst = ei + N_EDGES;    // edge_index[1]

  float* aggr = (float*)d_ws;                            // [N_NODES, 128]
  float* h    = aggr + (size_t)N_NODES * HID_C;          // [N_NODES, 128]

  const int n4 = N_NODES * HID_C / 4;                    // 3.2M float4s
  const int fill_grid = (n4 + 255) / 256;
  const int scat_grid = N_EDGES / 8;                     // 200000 (exact)
  const int gemm_grid = N_NODES / 16;                    // 6250 (exact)

  // ---- Layer 1 ----
  fill_neginf_kernel<<<fill_grid, 256, 0, stream>>>((float4*)aggr, n4);
  scatter_max_kernel<<<scat_grid, 256, 0, stream>>>(x, src, dst, aggr);
  // h = relu(fix(aggr) @ W1_l + b1 + x @ W1_r); N=128 -> 8 waves/block
  sage_gemm_kernel<IN_C, HID_C, true><<<gemm_grid, 256, 0, stream>>>(
      aggr, x, W1l, b1l, W1r, h);

  // ---- Layer 2 ----
  fill_neginf_kernel<<<fill_grid, 256, 0, stream>>>((float4*)aggr, n4);
  scatter_max_kernel<<<scat_grid, 256, 0, stream>>>(h, src, dst, aggr);
  // out = fix(aggr) @ W2_l + b2 + h @ W2_r; N=64 -> 4 waves/block
  sage_gemm_kernel<HID_C, OUT_C, false><<<gemm_grid, 128, 0, stream>>>(
      aggr, h, W2l, b2l, W2r, out);
}